// SimUnsimPrompt_58626303591036
// MI455X (gfx1250) — compile-verified
//
#include <hip/hip_runtime.h>
#include <hip/hip_bf16.h>
#include <math.h>

// Problem constants (from reference)
#define POOL 32
#define SEL 16
#define PLEN 8
#define DIM 768
#define BATCH 2048
#define EPS 1e-8f
#define TILE (PLEN * DIM)          // 6144 floats per prompt
#define SELTILE (SEL * TILE)       // 98304 floats = 384 KB

typedef __attribute__((ext_vector_type(2))) float v2f;
typedef __attribute__((ext_vector_type(8))) float v8f;
typedef __attribute__((ext_vector_type(4))) float f32x4;

// ---------------------------------------------------------------------------
// Kernel 1: row L2 norms of query (2048 rows) and prompt_key (32 rows),
// plus zeroing of the global histogram. One wave32 per row.
// Blocks 0..255 -> query rows (8 waves/block); blocks 256..259 -> key rows.
// ---------------------------------------------------------------------------
__global__ void norms_kernel(const float* __restrict__ query,
                             const float* __restrict__ pkey,
                             float* __restrict__ qn,
                             float* __restrict__ kn,
                             int* __restrict__ counts) {
    const int wave = threadIdx.x >> 5;
    const int lane = threadIdx.x & 31;
    if (blockIdx.x == 0 && threadIdx.x < POOL) counts[threadIdx.x] = 0;

    const float* src;
    int row;
    if (blockIdx.x < 256) {
        row = blockIdx.x * 8 + wave;          // 0..2047
        src = query + (size_t)row * DIM;
    } else {
        row = (blockIdx.x - 256) * 8 + wave;  // 0..31
        src = pkey + (size_t)row * DIM;
    }
    float s = 0.f;
    for (int i = lane; i < DIM; i += 32) {
        float v = src[i];
        s = fmaf(v, v, s);
    }
    #pragma unroll
    for (int off = 16; off > 0; off >>= 1) s += __shfl_down(s, off, 32);
    if (lane == 0) {
        if (blockIdx.x < 256) qn[row] = sqrtf(s);
        else                  kn[row] = sqrtf(s);
    }
}

// ---------------------------------------------------------------------------
// Kernel 2: match = (query @ prompt_key^T) / max(qn*kn, EPS)  via WMMA f32.
// One wave per 16x16 output tile: 128 M-tiles x 2 N-tiles = 256 waves
// = 32 blocks x 256 threads. K-loop of DIM/4 = 192 V_WMMA_F32_16X16X4_F32.
//
// A (16x4 f32) per ISA layout: lanes 0-15 hold row M=lane {K,K+1} in 2 VGPRs,
// lanes 16-31 hold {K+2,K+3}. B (4x16) mirrored over columns. C/D: VGPR r,
// lane L -> M = r + 8*(L>=16), N = L%16.
// ---------------------------------------------------------------------------
__global__ void match_wmma_kernel(const float* __restrict__ query,
                                  const float* __restrict__ pkey,
                                  const float* __restrict__ qn,
                                  const float* __restrict__ kn,
                                  float* __restrict__ match) {
    const int lane = threadIdx.x & 31;
    const int wave = blockIdx.x * (blockDim.x >> 5) + (threadIdx.x >> 5);
    const int m0 = (wave >> 1) << 4;   // 0..2032 step 16
    const int n0 = (wave & 1) << 4;    // 0 or 16
    const int half = lane >> 4;        // 0: lanes 0-15, 1: lanes 16-31
    const int l16 = lane & 15;

    const float* arow = query + (size_t)(m0 + l16) * DIM + (half << 1);
    const float* brow = pkey  + (size_t)(n0 + l16) * DIM + (half << 1);

    v8f acc = {};
    #pragma unroll 4
    for (int k = 0; k < DIM; k += 4) {
        v2f a = *(const v2f*)(arow + k);
        v2f b = *(const v2f*)(brow + k);
        acc = __builtin_amdgcn_wmma_f32_16x16x4_f32(
            /*neg_a=*/false, a, /*neg_b=*/false, b,
            /*c_mod=*/(short)0, acc, /*reuse_a=*/false, /*reuse_b=*/false);
    }

    const float knv = kn[n0 + l16];
    #pragma unroll
    for (int r = 0; r < 8; ++r) {
        const int M = m0 + r + (half << 3);
        float denom = fmaxf(qn[M] * knv, EPS);
        match[M * POOL + n0 + l16] = acc[r] / denom;
    }
}

// ---------------------------------------------------------------------------
// Kernel 3: per-row top-16 membership (rank-count, ties -> lower index, same
// as jax.lax.top_k) + histogram over pool ids. LDS hist per block, one
// global atomicAdd per pool id per block. 2048 threads = 8 blocks x 256.
// ---------------------------------------------------------------------------
__global__ void topk_hist_kernel(const float* __restrict__ match,
                                 int* __restrict__ counts) {
    __shared__ int lhist[POOL];
    if (threadIdx.x < POOL) lhist[threadIdx.x] = 0;
    __syncthreads();

    const int b = blockIdx.x * blockDim.x + threadIdx.x;   // < 2048
    float m[POOL];
    const float* row = match + b * POOL;
    #pragma unroll
    for (int i = 0; i < POOL; ++i) m[i] = row[i];

    for (int p = 0; p < POOL; ++p) {
        int rank = 0;
        #pragma unroll
        for (int q = 0; q < POOL; ++q)
            rank += (m[q] > m[p]) || ((m[q] == m[p]) && (q < p));
        if (rank < SEL) atomicAdd(&lhist[p], 1);
    }
    __syncthreads();
    if (threadIdx.x < POOL) atomicAdd(&counts[threadIdx.x], lhist[threadIdx.x]);
}

// ---------------------------------------------------------------------------
// Kernel 4: single block of 32 threads: mosts = top-16 pool ids by count
// (desc, ties -> lower id), nonk = non-selected ids ascending.
// ---------------------------------------------------------------------------
__global__ void select_kernel(const int* __restrict__ counts,
                              int* __restrict__ mosts,
                              int* __restrict__ nonk) {
    __shared__ int c[POOL];
    __shared__ int selflag[POOL];
    const int p = threadIdx.x;  // 0..31
    c[p] = counts[p];
    __syncthreads();
    int rank = 0;
    #pragma unroll
    for (int q = 0; q < POOL; ++q)
        rank += (c[q] > c[p]) || ((c[q] == c[p]) && (q < p));
    const int sel = (rank < SEL) ? 1 : 0;
    selflag[p] = sel;
    if (sel) mosts[rank] = p;
    __syncthreads();
    if (!sel) {
        int pos = 0;
        for (int q = 0; q < p; ++q) pos += (selflag[q] == 0);
        nonk[pos] = p;
    }
}

// ---------------------------------------------------------------------------
// Kernel 5: gather similarity / unsimilarity [2048,16] and the compact
// selected-prompts tile selp[16*8*768] (read once; broadcast later).
// 98304 threads = 384 blocks x 256.
// ---------------------------------------------------------------------------
__global__ void gather_kernel(const float* __restrict__ match,
                              const float* __restrict__ prompts,
                              const int* __restrict__ mosts,
                              const int* __restrict__ nonk,
                              float* __restrict__ sim,
                              float* __restrict__ unsim,
                              float* __restrict__ selp) {
    const int idx = blockIdx.x * blockDim.x + threadIdx.x;
    if (idx < BATCH * SEL) {
        const int b = idx >> 4, j = idx & 15;
        sim[idx]   = match[b * POOL + mosts[j]];
        unsim[idx] = match[b * POOL + nonk[j]];
    }
    if (idx < SELTILE) {
        const int j   = idx / TILE;
        const int rem = idx - j * TILE;
        selp[idx] = prompts[(size_t)mosts[j] * TILE + rem];
    }
}

// ---------------------------------------------------------------------------
// Kernel 6: the store-bound broadcast: selection[b] = selp for all b.
// 805 MB of B128 nontemporal stores; source (384 KB) stays resident in L2.
// grid = (96, 2048), 256 threads, one float4 per thread.
// ---------------------------------------------------------------------------
__global__ void broadcast_kernel(const f32x4* __restrict__ selp,
                                 f32x4* __restrict__ out_sel) {
    const int col = blockIdx.x * blockDim.x + threadIdx.x;  // 0..24575
    const size_t b = blockIdx.y;
    const f32x4 v = selp[col];
    __builtin_nontemporal_store(v, &out_sel[b * (SELTILE / 4) + col]);
}

// ---------------------------------------------------------------------------
extern "C" void kernel_launch(void* const* d_in, const int* in_sizes, int n_in,
                              void* d_out, int out_size, void* d_ws, size_t ws_size,
                              hipStream_t stream) {
    const float* query   = (const float*)d_in[0];   // [2048, 768]
    const float* pkey    = (const float*)d_in[1];   // [32, 768]
    const float* prompts = (const float*)d_in[2];   // [32, 8, 768]

    float* out = (float*)d_out;
    float* sim     = out;                       // [2048,16]
    float* unsim   = out + BATCH * SEL;         // [2048,16]
    float* out_sel = out + 2 * BATCH * SEL;     // [2048,16,8,768]

    // Workspace layout (256B aligned chunks)
    char* ws = (char*)d_ws;
    float* match  = (float*)(ws);                         // 2048*32 f32 = 256 KB
    float* qn     = (float*)(ws + 262144);                // 2048 f32
    float* kn     = (float*)(ws + 270336);                // 32 f32
    int*   counts = (int*)  (ws + 270592);                // 32 i32
    int*   mosts  = (int*)  (ws + 270848);                // 16 i32
    int*   nonk   = (int*)  (ws + 271104);                // 16 i32
    float* selp   = (float*)(ws + 271360);                // 98304 f32 = 384 KB

    // 1) norms (+ zero histogram)
    norms_kernel<<<dim3(260), dim3(256), 0, stream>>>(query, pkey, qn, kn, counts);

    // 2) cosine-similarity GEMM via V_WMMA_F32_16X16X4_F32
    match_wmma_kernel<<<dim3(32), dim3(256), 0, stream>>>(query, pkey, qn, kn, match);

    // 3) per-row top-16 -> histogram
    topk_hist_kernel<<<dim3(BATCH / 256), dim3(256), 0, stream>>>(match, counts);

    // 4) batchwise selection (mosts) and complement (nonk)
    select_kernel<<<dim3(1), dim3(POOL), 0, stream>>>(counts, mosts, nonk);

    // 5) gather sim/unsim and compact selected prompt tile
    gather_kernel<<<dim3(SELTILE / 256), dim3(256), 0, stream>>>(
        match, prompts, mosts, nonk, sim, unsim, selp);

    // 6) broadcast the 384 KB tile to all 2048 batch rows (805 MB of stores)
    broadcast_kernel<<<dim3(SELTILE / (4 * 256), BATCH), dim3(256), 0, stream>>>(
        (const f32x4*)selp, (f32x4*)out_sel);
}